// Mp_encoder_32968168964323
// MI455X (gfx1250) — compile-verified
//
#include <hip/hip_runtime.h>

#define H 128   // feature width (fixed by the model)

typedef __attribute__((ext_vector_type(2))) float v2f;   // A/B fragment: 16x4 f32 = 2 VGPRs
typedef __attribute__((ext_vector_type(8))) float v8f;   // C/D fragment: 16x16 f32 = 8 VGPRs

// ---------------------------------------------------------------------------
// Pair-interleaved transpose:  wt2[(k/2)*2H + 2*j + (k&1)] = W[j,k]
// => B-fragment (B[ka,j], B[ka+1,j]) for even ka is one aligned 8-byte load,
//    and lanes j..j+15 read a contiguous 128 B line.
// ---------------------------------------------------------------------------
__global__ void transpose_w_kernel(const float* __restrict__ w, float* __restrict__ wt2) {
    const int j = blockIdx.x;    // row of W == output column of s
    const int k = threadIdx.x;   // contraction index
    wt2[(k >> 1) * (2 * H) + 2 * j + (k & 1)] = w[j * H + k];
}

// ---------------------------------------------------------------------------
// s[n, j] = sum_k x[n,k] * W[j,k] + bias[j]
// One wave per 16-row strip, covering all 128 output columns (8 WMMA tiles).
// fp32 WMMA 16x16x4: A reused 8x per k-step; 32 k-steps; A and B fragments
// are each a single global_load_b64.
// ---------------------------------------------------------------------------
__global__ void __launch_bounds__(256) gemm_wmma_kernel(
        const float* __restrict__ x, const float* __restrict__ wt2,
        const float* __restrict__ bias, float* __restrict__ s, int n_rows)
{
    const int lane = threadIdx.x & 31;
    const int wave = (blockIdx.x * blockDim.x + threadIdx.x) >> 5;
    const int row0 = wave * 16;
    if (row0 >= n_rows) return;          // wave-uniform: EXEC stays all-1s for WMMA

    const int m  = lane & 15;            // M index (A) / N index (B, C/D)
    const int hi = lane >> 4;            // 0: lanes 0-15 (K=0,1), 1: lanes 16-31 (K=2,3)

    int rowm = row0 + m;
    if (rowm >= n_rows) rowm = n_rows - 1;     // clamp loads; stores guarded below
    const float* __restrict__ xrow = x + (long)rowm * H;

    v8f acc[8] = {};

    for (int k0 = 0; k0 < H; k0 += 4) {
        const int ka = k0 + 2 * hi;                       // even
        v2f a = *(const v2f*)(xrow + ka);                 // A VGPR0/1: K=ka, ka+1
        const float* __restrict__ wtk = wt2 + (ka >> 1) * (2 * H) + 2 * m;
#pragma unroll
        for (int t = 0; t < 8; ++t) {
            v2f b = *(const v2f*)(wtk + t * 32);          // B rows K=ka,ka+1 col t*16+m
            acc[t] = __builtin_amdgcn_wmma_f32_16x16x4_f32(
                /*neg_a=*/false, a, /*neg_b=*/false, b,
                /*c_mod=*/(short)0, acc[t], /*reuse_a=*/false, /*reuse_b=*/false);
        }
    }

    if (row0 + 16 <= n_rows) {
        // full tile: no per-store guards
#pragma unroll
        for (int t = 0; t < 8; ++t) {
            const int n  = t * 16 + m;
            const float bn = bias[n];
#pragma unroll
            for (int r = 0; r < 8; ++r) {
                const int rr = row0 + r + 8 * hi;  // D VGPR r: M=r (lo lanes), r+8 (hi lanes)
                s[(long)rr * H + n] = acc[t][r] + bn;
            }
        }
    } else {
#pragma unroll
        for (int t = 0; t < 8; ++t) {
            const int n  = t * 16 + m;
            const float bn = bias[n];
#pragma unroll
            for (int r = 0; r < 8; ++r) {
                const int rr = row0 + r + 8 * hi;
                if (rr < n_rows)
                    s[(long)rr * H + n] = acc[t][r] + bn;
            }
        }
    }
}

// ---------------------------------------------------------------------------
// out[n, :] = gcn_bias[:]   (so the scatter accumulates straight on top)
// ---------------------------------------------------------------------------
__global__ void init_bias_kernel(float* __restrict__ out, const float* __restrict__ bias,
                                 int n_rows) {
    const int gid = blockIdx.x * blockDim.x + threadIdx.x;  // n_rows*32 threads
    const int n = gid >> 5;
    const int q = gid & 31;
    if (n >= n_rows) return;
    const float4 b = *(const float4*)(bias + q * 4);
    *(float4*)(out + (long)n * H + q * 4) = b;
}

// ---------------------------------------------------------------------------
// out[row[e], :] += w[e] * s[col[e], :]   — one wave per edge, float4 per lane,
// L2-resident f32 atomics (both s and out fit in the 192 MB L2).
// ---------------------------------------------------------------------------
__global__ void scatter_kernel(const float* __restrict__ s, const int* __restrict__ erow,
                               const int* __restrict__ ecol, const float* __restrict__ ew,
                               float* __restrict__ out, int n_edges) {
    const int gid = blockIdx.x * blockDim.x + threadIdx.x;
    const int e = gid >> 5;
    const int q = gid & 31;
    if (e >= n_edges) return;
    const int   r = erow[e];          // wave-uniform loads (scalarized by compiler)
    const int   c = ecol[e];
    const float w = ew[e];
    const float4 v = *(const float4*)(s + (long)c * H + q * 4);  // contiguous 512B/wave
    float* dst = out + (long)r * H + q * 4;
    atomicAdd(dst + 0, v.x * w);
    atomicAdd(dst + 1, v.y * w);
    atomicAdd(dst + 2, v.z * w);
    atomicAdd(dst + 3, v.w * w);
}

// ---------------------------------------------------------------------------
// x = (x > 0) ? x : a * x   (PReLU), float4 vectorized, in place.
// ---------------------------------------------------------------------------
__global__ void prelu_kernel(float* __restrict__ v, const float* __restrict__ pa, int n_vec4) {
    const int gid = blockIdx.x * blockDim.x + threadIdx.x;
    if (gid >= n_vec4) return;
    const float a = pa[0];
    float4 t = ((float4*)v)[gid];
    t.x = t.x > 0.f ? t.x : a * t.x;
    t.y = t.y > 0.f ? t.y : a * t.y;
    t.z = t.z > 0.f ? t.z : a * t.z;
    t.w = t.w > 0.f ? t.w : a * t.w;
    ((float4*)v)[gid] = t;
}

// ---------------------------------------------------------------------------
// Inputs: h[N,H], edge_row[E], edge_col[E], edge_w[E],
//         fc_w[2,H,H], fc_b[2,H], gcn_bias[2,H], prelu_a[2]
// Workspace: s[N*H] | mid[N*H] | wt2[H*H]  (~102.5 MB)
// ---------------------------------------------------------------------------
extern "C" void kernel_launch(void* const* d_in, const int* in_sizes, int n_in,
                              void* d_out, int out_size, void* d_ws, size_t ws_size,
                              hipStream_t stream) {
    const float* h    = (const float*)d_in[0];
    const int*   erow = (const int*)  d_in[1];
    const int*   ecol = (const int*)  d_in[2];
    const float* ew   = (const float*)d_in[3];
    const float* fcw  = (const float*)d_in[4];
    const float* fcb  = (const float*)d_in[5];
    const float* gb   = (const float*)d_in[6];
    const float* pa   = (const float*)d_in[7];

    const int N = in_sizes[0] / H;
    const int E = in_sizes[1];

    float* out = (float*)d_out;
    float* s   = (float*)d_ws;
    float* mid = s   + (size_t)N * H;
    float* wt2 = mid + (size_t)N * H;

    const int gemm_blocks = ((N + 15) / 16 + 7) / 8;        // 8 waves (256 thr) per block
    const int nh_blocks   = (N * 32 + 255) / 256;           // N*32 threads (float4 per lane)
    const int e_blocks    = (E * 32 + 255) / 256;           // one wave per edge

    // ---- layer 0 ----
    transpose_w_kernel<<<H, H, 0, stream>>>(fcw, wt2);
    gemm_wmma_kernel  <<<gemm_blocks, 256, 0, stream>>>(h, wt2, fcb, s, N);
    init_bias_kernel  <<<nh_blocks, 256, 0, stream>>>(mid, gb, N);
    scatter_kernel    <<<e_blocks, 256, 0, stream>>>(s, erow, ecol, ew, mid, E);
    prelu_kernel      <<<nh_blocks, 256, 0, stream>>>(mid, pa, N * 32);

    // ---- layer 1 ----
    transpose_w_kernel<<<H, H, 0, stream>>>(fcw + H * H, wt2);
    gemm_wmma_kernel  <<<gemm_blocks, 256, 0, stream>>>(mid, wt2, fcb + H, s, N);
    init_bias_kernel  <<<nh_blocks, 256, 0, stream>>>(out, gb + H, N);
    scatter_kernel    <<<e_blocks, 256, 0, stream>>>(s, erow, ecol, ew, out, E);
    prelu_kernel      <<<nh_blocks, 256, 0, stream>>>(out, pa + 1, N * 32);
}